// HierarchicalCubeMap_41308995453180
// MI455X (gfx1250) — compile-verified
//
#include <hip/hip_runtime.h>
#include <cstdint>
#include <cstddef>

// ---------------------------------------------------------------------------
// HierarchicalCubeMap for MI455X (gfx1250).
// Phase A: build 9 mip levels in d_ws (2x2 box filter), staging the two input
//          rows per output row into LDS with the Tensor Data Mover (TDM).
// Phase B: one thread per ray: face/uv, mip select, 2x bilinear taps,
//          exp + clip. Non-temporal for streamed data, RT for L2-hot texels.
// ---------------------------------------------------------------------------

#define CUBE_R 512
#define N_MIPS 10

// Float offsets of mip levels 1..9 inside d_ws (level 0 is the input bg_mat).
// size(level m) = 6*3*(512>>m)^2
__device__ const long long kLvlOffDev[N_MIPS] = {
    0,        0,        1179648, 1474560, 1548288,
    1566720,  1571328,  1572480, 1572768, 1572840};

#if defined(__HIP_DEVICE_COMPILE__) && defined(__gfx1250__) && \
    __has_builtin(__builtin_amdgcn_tensor_load_to_lds) &&       \
    __has_builtin(__builtin_amdgcn_s_wait_tensorcnt)
#define MIP_USE_TDM 1
typedef unsigned int tdm_u32x4 __attribute__((ext_vector_type(4)));
typedef int          tdm_i32x8 __attribute__((ext_vector_type(8)));
typedef int          tdm_i32x4 __attribute__((ext_vector_type(4)));
#else
#define MIP_USE_TDM 0
#endif

// ---------------------------------------------------------------------------
// Mip downsample: one workgroup per output row (f*h + y). Input rows
// (f, 2y) and (f, 2y+1) are contiguous spans of W = 3*(2h) floats starting at
// global row 2*gr (faces tile evenly since r = 2h). TDM loads the 2 x W tile
// into LDS; threads then average 2x2 texel quads per channel.
// ---------------------------------------------------------------------------
__global__ __launch_bounds__(256) void mip_down_kernel(
    const float* __restrict__ in, float* __restrict__ out, int h) {
  const int r  = h << 1;
  const int W  = 3 * r;  // input row width in floats (<= 1536)
  const int Wo = 3 * h;  // output row width in floats
  __shared__ float rows[2 * 3 * CUBE_R];  // 12 KB max (2 rows of level-0)

  const int    gr      = blockIdx.x;                     // [0, 6h)
  const size_t srcBase = (size_t)(2 * gr) * (size_t)W;   // first input float

#if MIP_USE_TDM
  if (threadIdx.x < 32) {  // wave 0 issues the TDM descriptor (wave-level op)
    const unsigned long long ga  = (unsigned long long)(in + srcBase);
    const unsigned           lds = (unsigned)(unsigned long long)(&rows[0]);
    const unsigned uw = (unsigned)W;
    tdm_u32x4 g0;
    g0[0] = 1u;                                        // count=1, user D#
    g0[1] = lds;                                       // lds_addr (bytes)
    g0[2] = (unsigned)(ga & 0xFFFFFFFFull);            // global_addr[31:0]
    g0[3] = (unsigned)((ga >> 32) & 0x01FFFFFFull)     // global_addr[56:32]
            | 0x80000000u;                             // type = 2 ("image")
    tdm_i32x8 g1;
    g1[0] = (int)(2u << 16);                           // data_size = 4 bytes
    g1[1] = (int)((uw & 0xFFFFu) << 16);               // tensor_dim0[15:0]
    g1[2] = (int)(((uw >> 16) & 0xFFFFu) | (2u << 16));// td0[31:16]|td1=2 rows
    g1[3] = (int)((uw & 0xFFFFu) << 16);               // td1 hi=0 | tile_dim0=W
    g1[4] = 2;                                         // tile_dim1=2, tile_dim2=0
    g1[5] = (int)uw;                                   // dim0_stride[31:0] = W
    g1[6] = 0;                                         // dim0_stride hi | dim1_stride lo
    g1[7] = 0;
    tdm_i32x4 g2 = {0, 0, 0, 0};                       // 2D tensor: groups 2/3 unused
    tdm_i32x4 g3 = {0, 0, 0, 0};
    tdm_i32x8 g4 = {0, 0, 0, 0, 0, 0, 0, 0};           // unused extension group
    __builtin_amdgcn_tensor_load_to_lds(g0, g1, g2, g3, g4, 0);
    __builtin_amdgcn_s_wait_tensorcnt(0);
  }
  __syncthreads();
#else
  for (int i = threadIdx.x; i < 2 * W; i += 256) rows[i] = in[srcBase + i];
  __syncthreads();
#endif

  const size_t dstBase = (size_t)gr * (size_t)Wo;
  for (int e = threadIdx.x; e < Wo; e += 256) {
    const int xq = e / 3;
    const int c  = e - 3 * xq;
    const int j0 = 6 * xq + c;  // (2x)*3 + c
    const float a = rows[j0] + rows[j0 + 3] + rows[W + j0] + rows[W + j0 + 3];
    out[dstBase + e] = 0.25f * a;
  }
}

// ---------------------------------------------------------------------------
// Bilinear tap at one mip level (clamp-to-edge, matches reference clip order).
// ---------------------------------------------------------------------------
__device__ inline void sample_level(const float* __restrict__ tex, int R,
                                    int face, float s, float t, float rgb[3]) {
  const float u  = s * (float)R - 0.5f;
  const float v  = t * (float)R - 0.5f;
  const float uf = floorf(u), vf = floorf(v);
  const float fu = u - uf,    fv = v - vf;
  const int u0 = (int)uf, v0 = (int)vf;
  const int u0i = min(max(u0, 0), R - 1);
  const int u1i = min(max(u0 + 1, 0), R - 1);
  const int v0i = min(max(v0, 0), R - 1);
  const int v1i = min(max(v0 + 1, 0), R - 1);
  const size_t fb = (size_t)face * (size_t)R * (size_t)R;
  const float* p00 = tex + (fb + (size_t)v0i * R + u0i) * 3;
  const float* p01 = tex + (fb + (size_t)v0i * R + u1i) * 3;
  const float* p10 = tex + (fb + (size_t)v1i * R + u0i) * 3;
  const float* p11 = tex + (fb + (size_t)v1i * R + u1i) * 3;
  const float w00 = (1.0f - fu) * (1.0f - fv);
  const float w01 = fu * (1.0f - fv);
  const float w10 = (1.0f - fu) * fv;
  const float w11 = fu * fv;
#pragma unroll
  for (int c = 0; c < 3; ++c)
    rgb[c] = p00[c] * w00 + p01[c] * w01 + p10[c] * w10 + p11[c] * w11;
}

// ---------------------------------------------------------------------------
// Main sampling kernel: one thread per ray.
// ---------------------------------------------------------------------------
__global__ __launch_bounds__(256) void cubemap_sample_kernel(
    const float* __restrict__ vd, const float* __restrict__ sa,
    const float* __restrict__ lvl0, const float* __restrict__ ws,
    const float* __restrict__ pBright, const float* __restrict__ pMul,
    const float* __restrict__ pMipBias, float* __restrict__ out, int B) {
  const int i = blockIdx.x * blockDim.x + threadIdx.x;
  if (i >= B) return;

  // Streamed once -> non-temporal so L2 stays reserved for the texel pyramid.
  const float x   = __builtin_nontemporal_load(&vd[3 * i + 0]);
  const float y   = __builtin_nontemporal_load(&vd[3 * i + 1]);
  const float z   = __builtin_nontemporal_load(&vd[3 * i + 2]);
  const float saS = __builtin_nontemporal_load(&sa[i]);

  const float brightness = pBright[0];
  const float mul        = pMul[0];
  const float mipbias    = pMipBias[0];

  const float ax = fabsf(x), ay = fabsf(y), az = fabsf(z);
  const bool  is_x = (ax >= ay) && (ax >= az);
  const bool  is_y = !is_x && (ay >= az);
  const float ma = fmaxf(ax, fmaxf(ay, az));

  const int   face = is_x ? (x > 0.0f ? 0 : 1)
                          : (is_y ? (y > 0.0f ? 2 : 3) : (z > 0.0f ? 4 : 5));
  const float sc = is_x ? (x > 0.0f ? -z : z)
                        : (is_y ? x : (z > 0.0f ? x : -x));
  const float tc = is_x ? -y : (is_y ? (y > 0.0f ? z : -z) : -y);
  const float s = 0.5f * (sc / ma + 1.0f);
  const float t = 0.5f * (tc / ma + 1.0f);

  // mip = (sa - log(max(saTexel, EPS))) / log(4) / 2 + mipbias, clipped.
  const float saTexel = (1.0f / ma) / (float)CUBE_R / (float)CUBE_R;
  const float clipped = fmaxf(saTexel, 1.1920929e-07f);  // FLT_EPSILON
  float mip = (saS - logf(clipped)) * 0.3606737602222409f + mipbias; // 1/(2 ln 4)
  mip = fminf(fmaxf(mip, 0.0f), (float)(N_MIPS - 1));

  const int   m0   = (int)floorf(mip);
  const int   m1   = min(m0 + 1, N_MIPS - 1);
  const float frac = mip - (float)m0;

  float lo[3], hi[3];
  const float* t0 = (m0 == 0) ? lvl0 : (ws + kLvlOffDev[m0]);
  sample_level(t0, CUBE_R >> m0, face, s, t, lo);
  const float* t1 = (m1 == 0) ? lvl0 : (ws + kLvlOffDev[m1]);
  sample_level(t1, CUBE_R >> m1, face, s, t, hi);

  const float br = fminf(fmaxf(brightness, -1.0f), 2.0f);
#pragma unroll
  for (int c = 0; c < 3; ++c) {
    const float e = br + mul * (lo[c] * (1.0f - frac) + hi[c] * frac);
    const float o = fminf(fmaxf(expf(e), 0.01f), 1000.0f);
    __builtin_nontemporal_store(o, &out[3 * i + c]);
  }
}

// ---------------------------------------------------------------------------
extern "C" void kernel_launch(void* const* d_in, const int* in_sizes, int n_in,
                              void* d_out, int out_size, void* d_ws,
                              size_t ws_size, hipStream_t stream) {
  const float* vd   = (const float*)d_in[0];
  const float* sa   = (const float*)d_in[1];
  const float* bg   = (const float*)d_in[2];
  const float* pB   = (const float*)d_in[3];
  const float* pM   = (const float*)d_in[4];
  const float* pMip = (const float*)d_in[5];
  float*       out  = (float*)d_out;
  float*       ws   = (float*)d_ws;

  const int B = in_sizes[0] / 3;

  static const long long kLvlOffHost[N_MIPS] = {
      0,        0,        1179648, 1474560, 1548288,
      1566720,  1571328,  1572480, 1572768, 1572840};

  // Phase A: build mip levels 1..9, each from the previous level.
  const float* prev = bg;
  for (int m = 1; m < N_MIPS; ++m) {
    const int h   = CUBE_R >> m;
    float*    dst = ws + kLvlOffHost[m];
    mip_down_kernel<<<dim3(6 * h), dim3(256), 0, stream>>>(prev, dst, h);
    prev = dst;
  }

  // Phase B: sample all rays.
  const int grid = (B + 255) / 256;
  cubemap_sample_kernel<<<dim3(grid), dim3(256), 0, stream>>>(
      vd, sa, bg, ws, pB, pM, pMip, out, B);
}